// QueryImageCrossAttention_64338610094965
// MI455X (gfx1250) — compile-verified
//
#include <hip/hip_runtime.h>
#include <math.h>

// Problem constants (from reference)
#define NLVL   3
#define NPTS   4
#define NHEAD  32
#define HDIM   32
#define NQ_    64
#define P_     576
#define T_     640
#define D_     1024
#define LV_    1344
#define BSZ_   16
#define NEG_F  (-3.402823466e38f)

typedef float v2f __attribute__((ext_vector_type(2)));
typedef float v8f __attribute__((ext_vector_type(8)));

// ---------------------------------------------------------------------------
// GEMM: C[M,N] = A[M,K] @ B[K,N] + bias[N]   (f32, row-major)
// Block = 8 waves, block tile 64x64, K-chunk 16, double-buffered LDS staged
// with GLOBAL_LOAD_ASYNC_TO_LDS_B128 (ASYNCcnt) per CDNA5 ISA.
// Each wave computes two 16x16 tiles with V_WMMA_F32_16X16X4_F32.
// Requires: M % 64 == 0, N % 64 == 0, K % 16 == 0 (true for all launches).
// ---------------------------------------------------------------------------
__global__ __launch_bounds__(256) void gemm_bias_f32_wmma_lds(
    const float* __restrict__ A, const float* __restrict__ B,
    const float* __restrict__ bias, float* __restrict__ C,
    int M, int N, int K)
{
  __shared__ __align__(16) float ldsA[2][64 * 16];   // [buf][row*16 + k]
  __shared__ __align__(16) float ldsB[2][16 * 64];   // [buf][k*64 + n]

  const int tid  = threadIdx.x;
  const int wave = tid >> 5;
  const int lane = tid & 31;
  const int half = lane >> 4;        // K-pair select {0,1} vs {2,3}
  const int lr   = lane & 15;

  const int nBlocksN = N >> 6;
  const int bM = blockIdx.x / nBlocksN;
  const int bN = blockIdx.x - bM * nBlocksN;
  const int M0b = bM << 6;
  const int N0b = bN << 6;

  const int wr = wave >> 1;          // wave row tile 0..3  (16 rows each)
  const int wc = wave & 1;           // wave col group 0..1 (32 cols each)

  // Async staging assignment: one 16B segment of A and of B per thread/chunk.
  const int arow = tid >> 2;              // 0..63
  const int aseg = (tid & 3) << 2;        // float offset 0,4,8,12 within K-chunk
  const int brow = tid >> 4;              // 0..15 (k within chunk)
  const int bseg = (tid & 15) << 2;       // float offset 0..60 within 64 cols

  const float* Ag = A + (size_t)(M0b + arow) * K + aseg;
  const float* Bg = B + (size_t)brow * N + (N0b + bseg);

  const unsigned ldsAoff = (unsigned)(uintptr_t)(&ldsA[0][arow * 16 + aseg]);
  const unsigned ldsBoff = (unsigned)(uintptr_t)(&ldsB[0][brow * 64 + bseg]);
  const unsigned bufStrideA = (unsigned)(sizeof(float) * 64 * 16);
  const unsigned bufStrideB = (unsigned)(sizeof(float) * 16 * 64);

  v8f acc0 = {}, acc1 = {};

  const int nChunks = K >> 4;

  // Issue async copies of K-chunk `kc` into buffer `buf`.
  auto issue_chunk = [&](int buf, int kc) {
    unsigned la = ldsAoff + (unsigned)buf * bufStrideA;
    unsigned long long ga = (unsigned long long)(uintptr_t)(Ag + kc);
    asm volatile("global_load_async_to_lds_b128 %0, %1, off"
                 :: "v"(la), "v"(ga) : "memory");
    unsigned lb = ldsBoff + (unsigned)buf * bufStrideB;
    unsigned long long gb = (unsigned long long)(uintptr_t)(Bg + (size_t)kc * N);
    asm volatile("global_load_async_to_lds_b128 %0, %1, off"
                 :: "v"(lb), "v"(gb) : "memory");
  };

  issue_chunk(0, 0);

  for (int c = 0; c < nChunks; ++c) {
    // Own async loads complete -> barrier -> whole chunk resident for all waves.
    asm volatile("s_wait_asynccnt 0x0" ::: "memory");
    __syncthreads();

    if (c + 1 < nChunks) issue_chunk((c + 1) & 1, (c + 1) << 4);

    const float* As = &ldsA[c & 1][(wr * 16 + lr) * 16 + 2 * half];
    const float* Bs = &ldsB[c & 1][wc * 32 + lr];
#pragma unroll
    for (int ks = 0; ks < 4; ++ks) {
      const int kl = ks * 4 + 2 * half;         // local k of this half-wave
      v2f a = *(const v2f*)(As + ks * 4);       // ds_load_b64
      v2f b0, b1;
      b0.x = Bs[(size_t)kl * 64];
      b0.y = Bs[(size_t)(kl + 1) * 64];
      b1.x = Bs[(size_t)kl * 64 + 16];
      b1.y = Bs[(size_t)(kl + 1) * 64 + 16];
      acc0 = __builtin_amdgcn_wmma_f32_16x16x4_f32(
          false, a, false, b0, (short)0, acc0, false, false);
      acc1 = __builtin_amdgcn_wmma_f32_16x16x4_f32(
          false, a, false, b1, (short)0, acc1, false, false);
    }
    __syncthreads();   // compute(c) done before buffer is re-issued at c+1
  }

  const int rowBase = M0b + wr * 16 + 8 * half;
  const int col0 = N0b + wc * 32 + lr;
  const float bv0 = bias[col0];
  const float bv1 = bias[col0 + 16];
  float* Cp = C + (size_t)rowBase * N + col0;
#pragma unroll
  for (int j = 0; j < 8; ++j) {
    Cp[(size_t)j * N]      = acc0[j] + bv0;
    Cp[(size_t)j * N + 16] = acc1[j] + bv1;
  }
}

// ---------------------------------------------------------------------------
// Softmax over groups of 12 (NL*NPT) per (b, q, head), in place.
// logits = aw + (1 - mask[b]) * NEG  (reproduces reference f32 saturation)
// ---------------------------------------------------------------------------
__global__ __launch_bounds__(256) void softmax12_kernel(
    float* __restrict__ aw, const float* __restrict__ mask)
{
  const int idx = blockIdx.x * blockDim.x + threadIdx.x;   // (b*NQ)*NH + h
  if (idx >= BSZ_ * NQ_ * NHEAD) return;
  const int h   = idx & (NHEAD - 1);
  const int row = idx / NHEAD;                             // b*NQ + q
  const int b   = row / NQ_;

  const float bias = (1.0f - mask[b]) * NEG_F;
  float* p = aw + (size_t)row * (NHEAD * 12) + h * 12;

  float v[12];
  float m = -INFINITY;
#pragma unroll
  for (int j = 0; j < 12; ++j) { v[j] = p[j] + bias; m = fmaxf(m, v[j]); }
  float s = 0.0f;
#pragma unroll
  for (int j = 0; j < 12; ++j) { v[j] = expf(v[j] - m); s += v[j]; }
  const float inv = 1.0f / s;
#pragma unroll
  for (int j = 0; j < 12; ++j) { p[j] = v[j] * inv; }
}

// ---------------------------------------------------------------------------
// Deformable bilinear sampling. One wave per (b, q, head); lane = head-dim d.
// value: (B, LV, NH, HD)  aw: (B, NQ, NH, 12)  off: (B, NQ, NH*NL*NPT*2)
// refp: (B, T, NL, 2)     out: (B, NQ, NH*HD)
// ---------------------------------------------------------------------------
__global__ __launch_bounds__(256) void deform_sample_kernel(
    const float* __restrict__ value, const float* __restrict__ aw,
    const float* __restrict__ off,   const float* __restrict__ refp,
    float* __restrict__ out)
{
  const int wave = threadIdx.x >> 5;
  const int lane = threadIdx.x & 31;          // d in [0, HD)
  const int unit = blockIdx.x * 8 + wave;     // (b*NQ + q)*NH + h
  if (unit >= BSZ_ * NQ_ * NHEAD) return;

  const int h  = unit & (NHEAD - 1);
  const int bq = unit >> 5;                   // b*NQ + q
  const int b  = bq / NQ_;
  const int q  = bq - b * NQ_;
  const int t  = P_ + q;

  const int lvDim[3]   = {32, 16, 8};
  const int lvStart[3] = {0, 1024, 1280};

  const float* awp  = aw  + ((size_t)bq * NHEAD + h) * 12;
  const float* offp = off + (size_t)bq * (NHEAD * NLVL * NPTS * 2) + h * (NLVL * NPTS * 2);
  const float* refb = refp + ((size_t)b * T_ + t) * (NLVL * 2);

  float acc = 0.0f;
#pragma unroll
  for (int l = 0; l < NLVL; ++l) {
    const int   HW  = lvDim[l];               // square levels: H_ == W_
    const float fHW = (float)HW;
    const float rx = refb[l * 2 + 0];
    const float ry = refb[l * 2 + 1];
    const float* vbase =
        value + (((size_t)b * LV_ + lvStart[l]) * NHEAD + h) * HDIM + lane;

#pragma unroll
    for (int p = 0; p < NPTS; ++p) {
      const float w  = awp[l * NPTS + p];
      const float ox = offp[l * (NPTS * 2) + p * 2 + 0];
      const float oy = offp[l * (NPTS * 2) + p * 2 + 1];
      const float x = (rx + ox / fHW) * fHW - 0.5f;
      const float y = (ry + oy / fHW) * fHW - 0.5f;
      const float x0f = floorf(x), y0f = floorf(y);
      const float fx = x - x0f, fy = y - y0f;
      const int x0 = (int)x0f, y0 = (int)y0f;

      float s = 0.0f;
#pragma unroll
      for (int dy = 0; dy < 2; ++dy) {
#pragma unroll
        for (int dx = 0; dx < 2; ++dx) {
          const int xi = x0 + dx, yi = y0 + dy;
          const float tw = (dx ? fx : 1.0f - fx) * (dy ? fy : 1.0f - fy);
          const bool valid = (xi >= 0) & (xi < HW) & (yi >= 0) & (yi < HW);
          const int xc = min(max(xi, 0), HW - 1);
          const int yc = min(max(yi, 0), HW - 1);
          const float vv = vbase[(size_t)(yc * HW + xc) * (NHEAD * HDIM)];
          s += vv * (valid ? tw : 0.0f);
        }
      }
      acc += w * s;
    }
  }
  out[(size_t)bq * (NHEAD * HDIM) + h * HDIM + lane] = acc;
}

// ---------------------------------------------------------------------------
// Host launcher
// ---------------------------------------------------------------------------
extern "C" void kernel_launch(void* const* d_in, const int* in_sizes, int n_in,
                              void* d_out, int out_size, void* d_ws, size_t ws_size,
                              hipStream_t stream) {
  (void)in_sizes; (void)n_in; (void)out_size; (void)ws_size;
  // Prompt-path inputs (0,4,6,7) are dead w.r.t. out[:, P:, :]; unused.
  const float* bypass = (const float*)d_in[1];   // (16, 64, 1024)
  const float* feats  = (const float*)d_in[2];   // (16, 1, 1344, 1024)
  const float* mask   = (const float*)d_in[3];   // (16, 1)
  const float* refp   = (const float*)d_in[5];   // (16, 640, 3, 2)
  const float* W_off  = (const float*)d_in[8];   // (1024, 768)
  const float* b_off  = (const float*)d_in[9];
  const float* W_attw = (const float*)d_in[10];  // (1024, 384)
  const float* b_attw = (const float*)d_in[11];
  const float* W_v    = (const float*)d_in[12];  // (1024, 1024)
  const float* b_v    = (const float*)d_in[13];
  float* out = (float*)d_out;                    // (16, 64, 1024)

  // Workspace layout (f32): value | off | aw  -> ~92.8 MB
  float* val  = (float*)d_ws;                    // 16*1344*1024
  float* offb = val + (size_t)BSZ_ * LV_ * D_;
  float* awb  = offb + (size_t)BSZ_ * NQ_ * (NHEAD * NLVL * NPTS * 2);

  // 1) value = feats @ W_v + b_v        M=21504 N=1024 K=1024
  {
    const int M = BSZ_ * LV_, N = D_, K = D_;
    gemm_bias_f32_wmma_lds<<<(M / 64) * (N / 64), 256, 0, stream>>>(
        feats, W_v, b_v, val, M, N, K);
  }
  // 2) off = bypass @ W_off + b_off     M=1024 N=768 K=1024
  {
    const int M = BSZ_ * NQ_, N = NHEAD * NLVL * NPTS * 2, K = D_;
    gemm_bias_f32_wmma_lds<<<(M / 64) * (N / 64), 256, 0, stream>>>(
        bypass, W_off, b_off, offb, M, N, K);
  }
  // 3) aw = bypass @ W_attw + b_attw    M=1024 N=384 K=1024
  {
    const int M = BSZ_ * NQ_, N = NHEAD * NLVL * NPTS, K = D_;
    gemm_bias_f32_wmma_lds<<<(M / 64) * (N / 64), 256, 0, stream>>>(
        bypass, W_attw, b_attw, awb, M, N, K);
  }
  // 4) softmax over NL*NPT=12 per (b,q,h), with image-mask bias
  {
    const int n = BSZ_ * NQ_ * NHEAD;   // 32768
    softmax12_kernel<<<(n + 255) / 256, 256, 0, stream>>>(awb, mask);
  }
  // 5) deformable bilinear sampling -> out
  {
    const int units = BSZ_ * NQ_ * NHEAD;   // 32768 waves
    deform_sample_kernel<<<(units + 7) / 8, 256, 0, stream>>>(
        val, awb, offb, refp, out);
  }
}